// QuantumAutoencoder_32925219291471
// MI455X (gfx1250) — compile-verified
//
#include <hip/hip_runtime.h>
#include <hip/hip_bf16.h>

// Quantum autoencoder forward on MI455X (gfx1250).
//
// Math: state stays REAL (RY + CZ are real, input real); <Y> single-wire terms
// are exactly 0; the circuit is a fixed 256x256 ORTHOGONAL matrix U (only the
// 26 weights), so the batch is one fp32 GEMM  Y = X * U^T  done with
// V_WMMA_F32_16X16X4_F32, followed by lane-local quadratic forms on groups of
// 4 amplitudes (wires 6/7 = the two LSBs of the amplitude index), scaled by
// 1/||x||^2 = 1/||y||^2 (U orthogonal => norm computed from Y directly).
//
// Roofline: GEMM 4.3 GFLOP; HBM traffic 32MB in + 1.2MB out ~= 1.5us @23.3TB/s.
// The real constraint is L2 re-reads of the 256KB packed W: with T=2 batch
// tiles per wave each B fragment feeds 2 WMMAs -> 256MB total L2 traffic
// (halved vs T=1). LDS is exactly 64KB: 32KB packed A-fragments + 32KB
// XOR-swizzled Y, both bank-conflict-free without padding.

#define NQ 8
#define DIM 256
#define BATCH 32768

typedef __attribute__((ext_vector_type(2))) float v2f;
typedef __attribute__((ext_vector_type(8))) float v8f;

// ---------------------------------------------------------------------------
// Kernel 1: build U by evolving the 256x256 identity through the circuit.
// Each block owns 64 input-basis columns (chunk = blockIdx.x), stored in LDS
// as S[row n][col c_local] (stride 64 words -> lanes hit distinct banks).
// Output: W = U^T packed in WMMA B-fragment order into d_ws:
//   Wf2[(kb*16+nb)*32 + lane] = { W[4kb+2*hi  ][16nb+lm],
//                                 W[4kb+2*hi+1][16nb+lm] },  hi=lane>>4, lm=lane&15
// matching the ISA 4x16 fp32 B layout (lanes 0-15: K=0,1; lanes 16-31: K=2,3).
// ---------------------------------------------------------------------------
__global__ void __launch_bounds__(1024) qae_build_u(const float* __restrict__ weights,
                                                    float* __restrict__ Wf) {
  __shared__ float S[256 * 64];              // 64 KB, one 64-column chunk of U
  const int tid   = threadIdx.x;
  const int chunk = blockIdx.x;              // which 64 columns (k-range)
  const int c     = tid & 63;                // local column (input basis idx)
  const int s16   = tid >> 6;                // 16-way split of row/pair work

  const int cg = 64 * chunk + c;             // identity init
  for (int n = 16 * s16; n < 16 * s16 + 16; ++n)
    S[n * 64 + c] = (n == cg) ? 1.0f : 0.0f;
  __syncthreads();

  for (int l = 0; l < 4; ++l) {
    const int nq = (l == 3) ? 2 : NQ;        // last layer: RY on trash wires only
    for (int qw = 0; qw < nq; ++qw) {
      const float th = 0.5f * weights[l * NQ + qw];
      const float ct = cosf(th), st = sinf(th);
      const int m = 1 << (7 - qw);           // wire w -> bit (7-w)  (wire0 = MSB)
      for (int p = 8 * s16; p < 8 * s16 + 8; ++p) {
        const int lo = p & (m - 1);
        const int n0 = ((p & ~(m - 1)) << 1) | lo;
        const int n1 = n0 | m;
        const float a = S[n0 * 64 + c];
        const float b = S[n1 * 64 + c];
        S[n0 * 64 + c] = ct * a - st * b;
        S[n1 * 64 + c] = st * a + ct * b;
      }
      __syncthreads();
    }
    if (l < 3) {
      // fused CZ layer: pairs (w0, w1..w7) and (w1, w2..w7)
      for (int n = 16 * s16; n < 16 * s16 + 16; ++n) {
        const int b0 = (n >> 7) & 1;
        const int b1 = (n >> 6) & 1;
        const int rest = __popc(n & 0x3F);
        const int e = b0 * (b1 + rest) + b1 * rest;
        if (e & 1) S[n * 64 + c] = -S[n * 64 + c];
      }
      __syncthreads();
    }
  }

  // pack this chunk's 16 kb-tiles of W = U^T in B-fragment order
  for (int i = 0; i < 8; ++i) {
    const int idx  = tid + 1024 * i;         // 0..8191 float2 per chunk
    const int lane = idx & 31;
    const int nb   = (idx >> 5) & 15;
    const int kbl  = idx >> 9;               // 0..15
    const int hi = lane >> 4, lm = lane & 15;
    const int k0l = 4 * kbl + 2 * hi;        // local k (column of this chunk)
    const int n   = 16 * nb + lm;            // output row index of U
    float2 v;
    v.x = S[n * 64 + k0l];                   // W[k][n] = U[n][k] = S[n][k_local]
    v.y = S[n * 64 + k0l + 1];
    const int kb = chunk * 16 + kbl;
    ((float2*)Wf)[(kb * 16 + nb) * 32 + lane] = v;
  }
}

// ---------------------------------------------------------------------------
// Kernel 2: one wave per block, T=2 batch tiles (32 rows) per wave so each
// B fragment (global, coalesced b64) feeds two WMMAs.
//  - Xa: A fragments pre-packed, entry (t*64+kb)*32 + (lane ^ (kb&31)).
//    The kb-XOR makes BOTH the staging stores (lanes span kb) and the GEMM
//    reads (lanes span lane-id) hit distinct LDS banks. Layout matches the
//    ISA 16x4 fp32 A tile: lanes 0-15 K=0,1 / lanes 16-31 K=2,3 of row lane&15.
//  - Ysw: D tiles scattered per the documented C/D layout (VGPR v, lane L ->
//    row v+8*(L>=16), col 16nb+(L&15)) at col ^ (4*row): conflict-free scatter
//    AND conflict-free aligned float4 row reads in the reduction.
// ---------------------------------------------------------------------------
__global__ void __launch_bounds__(32) qae_main(const float* __restrict__ X,
                                               const float* __restrict__ Wf,
                                               float* __restrict__ out) {
  __shared__ float2 Xa[2 * 64 * 32];         // 32 KB packed A fragments
  __shared__ float  Ysw[32 * 256];           // 32 KB swizzled Y tile
  const int lane = threadIdx.x;
  const int m0   = blockIdx.x * 32;

  // stage X: coalesced float4 reads, swizzled conflict-free fragment stores
  for (int i = 0; i < 64; ++i) {
    const int f   = lane + 32 * i;           // float4 index, 0..2047
    const int row = f >> 6;                  // 0..31
    const int kb  = f & 63;
    const float4 x = ((const float4*)X)[(size_t)(m0 + row) * 64 + kb];
    const int t = row >> 4, lm = row & 15;
    const int base = (t * 64 + kb) * 32;
    const int K = kb & 31;
    Xa[base + ((lm)      ^ K)] = make_float2(x.x, x.y);   // hi=0 half: K=0,1
    Xa[base + ((lm + 16) ^ K)] = make_float2(x.z, x.w);   // hi=1 half: K=2,3
  }
  __syncthreads();

  const int hi = lane >> 4, lm = lane & 15;
  const float2* __restrict__ W2 = (const float2*)Wf;

  for (int nb = 0; nb < 16; ++nb) {
    v8f acc0 = {}, acc1 = {};
#pragma unroll 4
    for (int kb = 0; kb < 64; ++kb) {
      const int lx = lane ^ (kb & 31);
      const v2f a0 = *(const v2f*)&Xa[kb * 32 + lx];            // tile t=0
      const v2f a1 = *(const v2f*)&Xa[(64 + kb) * 32 + lx];     // tile t=1
      const v2f b  = *(const v2f*)&W2[(kb * 16 + nb) * 32 + lane];
      acc0 = __builtin_amdgcn_wmma_f32_16x16x4_f32(
          false, a0, false, b, (short)0, acc0, false, false);
      acc1 = __builtin_amdgcn_wmma_f32_16x16x4_f32(
          false, a1, false, b, (short)0, acc1, false, false);
    }
    const int col = nb * 16 + lm;
#pragma unroll
    for (int v = 0; v < 8; ++v) {
      const int r0 = v + 8 * hi;             // tile 0 row
      const int r1 = r0 + 16;                // tile 1 row
      Ysw[r0 * 256 + (col ^ (4 * r0))] = acc0[v];
      Ysw[r1 * 256 + (col ^ (4 * r1))] = acc1[v];
    }
  }
  __syncthreads();

  // Expectations: wires 6/7 are index bits 1/0 -> groups of 4 columns.
  // U orthogonal => ||x||^2 == ||y||^2, so the norm comes from Y too.
  const int r = lane;                        // each lane reduces one row
  float nrm = 0.f, tx0 = 0.f, tz0 = 0.f, tx1 = 0.f, tz1 = 0.f;
  float txx = 0.f, tyy = 0.f, tzz = 0.f;
  for (int g = 0; g < 64; ++g) {
    const float4 y = *(const float4*)&Ysw[r * 256 + ((4 * g) ^ (4 * r))];
    nrm += y.x * y.x + y.y * y.y + y.z * y.z + y.w * y.w;
    tx0 += y.x * y.z + y.y * y.w;                                   // X on w6
    tz0 += (y.x * y.x + y.y * y.y) - (y.z * y.z + y.w * y.w);       // Z on w6
    tx1 += y.x * y.y + y.z * y.w;                                   // X on w7
    tz1 += y.x * y.x - y.y * y.y + y.z * y.z - y.w * y.w;           // Z on w7
    txx += y.x * y.w + y.y * y.z;                                   // X6 X7
    tyy += y.y * y.z - y.x * y.w;                                   // Y6 Y7
    tzz += y.x * y.x - y.y * y.y - y.z * y.z + y.w * y.w;           // Z6 Z7
  }
  const float inv = 1.0f / nrm;              // folds in AmplitudeEmbedding norm
  float* o = out + (size_t)(m0 + r) * 9;
  o[0] = 2.0f * tx0 * inv;
  o[1] = 0.0f;                               // <Y6> == 0 for real states
  o[2] = tz0 * inv;
  o[3] = 2.0f * tx1 * inv;
  o[4] = 0.0f;                               // <Y7> == 0 for real states
  o[5] = tz1 * inv;
  o[6] = 2.0f * txx * inv;
  o[7] = 2.0f * tyy * inv;
  o[8] = tzz * inv;
}

extern "C" void kernel_launch(void* const* d_in, const int* in_sizes, int n_in,
                              void* d_out, int out_size, void* d_ws, size_t ws_size,
                              hipStream_t stream) {
  const float* x = (const float*)d_in[0];    // [32768, 256] f32
  const float* w = (const float*)d_in[1];    // [26] f32
  float* out = (float*)d_out;                // [32768, 9] f32
  float* Wf  = (float*)d_ws;                 // 256 KB fragment-packed U^T

  qae_build_u<<<4, 1024, 0, stream>>>(w, Wf);
  qae_main<<<BATCH / 32, 32, 0, stream>>>(x, Wf, out);
}